// Router_41635412967985
// MI455X (gfx1250) — compile-verified
//
#include <hip/hip_runtime.h>
#include <math.h>

// ---------------------------------------------------------------------------
// MoE router for MI455X (gfx1250):
//   logits = x[32768,2048] @ W[2048,64]  (fp32, via V_WMMA_F32_16X16X4_F32)
//   softmax(64) -> top-8 -> softmax(8)
// Block: 256 threads (8 waves). Wave w handles tokens [block*128 + w*16, +16).
// K processed in 256-wide chunks; W chunk staged in LDS PRE-SWIZZLED into
// B-fragment order so each fragment is a single conflict-free ds_load_b64.
// ---------------------------------------------------------------------------

typedef __attribute__((ext_vector_type(2))) float v2f;
typedef __attribute__((ext_vector_type(8))) float v8f;

#define DDIM 2048
#define EXPERTS 64
#define KCHUNK 256
#define TOK_PER_BLOCK 128

__global__ __launch_bounds__(256) void router_wmma_kernel(
    const float* __restrict__ x,       // [tokens, 2048]
    const float* __restrict__ w,       // [2048, 64]
    float* __restrict__ out_w,         // [tokens, 8]
    int* __restrict__ out_i)           // [tokens, 8]
{
    // B-fragment-ordered W chunk: 64 K-groups x (4 nt-blocks x 32 lanes x 2 floats)
    // = 16384 floats = 64 KB. Reused for logits (32 KB) in the epilogue.
    __shared__ float lds[KCHUNK * EXPERTS];

    const int tid    = threadIdx.x;
    const int lane   = tid & 31;
    const int waveId = tid >> 5;
    const int blockBase = blockIdx.x * TOK_PER_BLOCK;
    const int tileBase  = blockBase + waveId * 16;

    // A-fragment addressing (16x4 fp32): lanes 0-15 -> K={0,1}, lanes 16-31 -> K={2,3}
    const int m     = lane & 15;
    const int halfK = (lane >> 4) * 2;    // 0 or 2
    const float* xrow = x + (size_t)(tileBase + m) * DDIM;

    v8f acc0 = {}, acc1 = {}, acc2 = {}, acc3 = {};

    for (int kc = 0; kc < DDIM; kc += KCHUNK) {
        __syncthreads();
        // Stage W[kc..kc+KCHUNK) x 64 into LDS in B-fragment order.
        // Pair p: k0g=p>>7, rem=p&127, nt=rem>>5, l=rem&31.
        //   e = nt*16 + (l&15), kb = k0g*4 + (l>>4)*2
        //   lds[2p+0] = W[kc+kb][e]; lds[2p+1] = W[kc+kb+1][e]
        for (int p = tid; p < (KCHUNK * EXPERTS) / 2; p += 256) {
            int k0g = p >> 7;
            int rem = p & 127;
            int nt  = rem >> 5;
            int l   = rem & 31;
            int e   = nt * 16 + (l & 15);
            int kb  = k0g * 4 + (l >> 4) * 2;
            const float* src = w + (size_t)(kc + kb) * EXPERTS + e;
            v2f pr = { src[0], src[EXPERTS] };         // K=kb, K=kb+1 (same expert)
            *(v2f*)&lds[(size_t)p * 2] = pr;           // sequential ds_store_b64
        }
        __syncthreads();

        const float* bptr = &lds[lane * 2];            // lane's pair base

        #pragma unroll 4
        for (int k0 = 0; k0 < KCHUNK; k0 += 4) {
            v2f a = *(const v2f*)(xrow + kc + k0 + halfK);       // global_load_b64
            const float* bp = bptr + (k0 >> 2) * 256;            // K-group base
            v2f b0 = *(const v2f*)(bp +   0);                    // ds_load_b64 x4
            v2f b1 = *(const v2f*)(bp +  64);
            v2f b2 = *(const v2f*)(bp + 128);
            v2f b3 = *(const v2f*)(bp + 192);
            acc0 = __builtin_amdgcn_wmma_f32_16x16x4_f32(false, a, false, b0, (short)0, acc0, false, false);
            acc1 = __builtin_amdgcn_wmma_f32_16x16x4_f32(false, a, false, b1, (short)0, acc1, false, false);
            acc2 = __builtin_amdgcn_wmma_f32_16x16x4_f32(false, a, false, b2, (short)0, acc2, false, false);
            acc3 = __builtin_amdgcn_wmma_f32_16x16x4_f32(false, a, false, b3, (short)0, acc3, false, false);
        }
    }

    __syncthreads();   // everyone done with W chunks; reuse LDS for logits

    // Scatter accumulators so token t (0..127 in block) occupies lds[t*64 .. t*64+63].
    // C/D layout: VGPR j, lanes 0-15 -> M=j, lanes 16-31 -> M=j+8; N = lane&15.
    {
        float* lb = &lds[waveId * 1024];      // 16 tokens * 64 experts per wave
        const int hi = lane >> 4;
        const int n  = lane & 15;
        #pragma unroll
        for (int j = 0; j < 8; ++j) {
            int mm = j + 8 * hi;
            lb[mm * 64 +  0 + n] = acc0[j];
            lb[mm * 64 + 16 + n] = acc1[j];
            lb[mm * 64 + 32 + n] = acc2[j];
            lb[mm * 64 + 48 + n] = acc3[j];
        }
    }
    __syncthreads();

    // Per-token: softmax(64) -> top-8 (lowest-index tie break) -> softmax(8)
    if (tid < TOK_PER_BLOCK) {
        const float* lg = &lds[tid * 64];

        float mx = -INFINITY;
        for (int i = 0; i < 64; ++i) mx = fmaxf(mx, lg[i]);

        float s = 0.0f;
        for (int i = 0; i < 64; ++i) s += expf(lg[i] - mx);
        const float inv_s = 1.0f / s;

        unsigned long long used = 0ull;
        float w8[8];
        int   id8[8];
        #pragma unroll
        for (int j = 0; j < 8; ++j) {
            float best = -INFINITY;
            int   bi   = 0;
            for (int i = 0; i < 64; ++i) {
                float v = lg[i];
                bool free_i = (((used >> i) & 1ull) == 0ull);
                if (free_i && v > best) { best = v; bi = i; }
            }
            used |= (1ull << bi);
            w8[j]  = expf(best - mx) * inv_s;   // gating prob of selected expert
            id8[j] = bi;
        }

        // Second softmax over the 8 selected gating probs (descending -> max = w8[0])
        float e8[8];
        float s2 = 0.0f;
        #pragma unroll
        for (int j = 0; j < 8; ++j) { e8[j] = expf(w8[j] - w8[0]); s2 += e8[j]; }
        const float inv_s2 = 1.0f / s2;

        const int token = blockBase + tid;
        float* ow = out_w + (size_t)token * 8;
        int*   oi = out_i + (size_t)token * 8;
        #pragma unroll
        for (int j = 0; j < 8; ++j) {
            ow[j] = e8[j] * inv_s2;
            oi[j] = id8[j];
        }
    }
}

extern "C" void kernel_launch(void* const* d_in, const int* in_sizes, int n_in,
                              void* d_out, int out_size, void* d_ws, size_t ws_size,
                              hipStream_t stream) {
    const float* x = (const float*)d_in[0];     // [4,8192,2048] fp32
    const float* w = (const float*)d_in[1];     // [2048,64] fp32

    const int tokens = in_sizes[0] / DDIM;      // 32768
    float* out_w = (float*)d_out;                       // weights first
    int*   out_i = (int*)d_out + (size_t)tokens * 8;    // then indices

    dim3 grid(tokens / TOK_PER_BLOCK);
    dim3 block(256);
    hipLaunchKernelGGL(router_wmma_kernel, grid, block, 0, stream,
                       x, w, out_w, out_i);
}